// LightGCN_10754598109945
// MI455X (gfx1250) — compile-verified
//
#include <hip/hip_runtime.h>

typedef float v2f __attribute__((ext_vector_type(2)));
typedef float v8f __attribute__((ext_vector_type(8)));

#define HID 32
#define IND 128

// ---------------------------------------------------------------------------
// Projection: Z[r,h] = sum_k X[r,k]*W[k,h] + b[h], via V_WMMA_F32_16X16X4_F32.
// One wave per 16x16 output tile. rows must be a multiple of 16 (100000/40000 are).
// ---------------------------------------------------------------------------
__global__ void proj_wmma_f32(const float* __restrict__ X,
                              const float* __restrict__ W,
                              const float* __restrict__ b,
                              float* __restrict__ Z, int rows) {
  int wave = (int)((blockIdx.x * blockDim.x + threadIdx.x) >> 5);
  int lane = (int)(threadIdx.x & 31);
  int tiles_m = rows >> 4;
  int total = tiles_m * (HID / 16);
  if (wave >= total) return;              // whole-wave uniform: EXEC stays all-1s

  int tm = wave >> 1;                     // 16-row tile
  int tn = wave & 1;                      // 16-col tile (HID=32 -> 2)
  int row0 = tm << 4;
  int col0 = tn << 4;
  int n = lane & 15;                      // N index for B/C, M index for A
  int hi = lane >> 4;                     // half-wave selector
  int koff = hi * 2;                      // A/B: lanes 0-15 hold K+{0,1}, 16-31 K+{2,3}

  float bias = b[col0 + n];
  v8f c;
#pragma unroll
  for (int j = 0; j < 8; ++j) c[j] = bias;

  const float* xrow = X + (size_t)(row0 + n) * IND;  // A: M = lane&15
#pragma unroll
  for (int kc = 0; kc < IND; kc += 4) {
    v2f a, bm;
    a[0] = xrow[kc + koff];
    a[1] = xrow[kc + koff + 1];
    bm[0] = W[(size_t)(kc + koff) * HID + col0 + n];
    bm[1] = W[(size_t)(kc + koff + 1) * HID + col0 + n];
    c = __builtin_amdgcn_wmma_f32_16x16x4_f32(
        /*neg_a=*/false, a, /*neg_b=*/false, bm,
        /*c_mod=*/(short)0, c, /*reuse_a=*/false, /*reuse_b=*/false);
  }

#pragma unroll
  for (int j = 0; j < 8; ++j) {
    Z[(size_t)(row0 + j + hi * 8) * HID + col0 + n] = c[j];
  }
}

// ---------------------------------------------------------------------------
// Degree accumulation: each edge contributes 1 to its user node and item node.
// ---------------------------------------------------------------------------
__global__ void degree_kernel(const int* __restrict__ eu, const int* __restrict__ ei,
                              float* __restrict__ deg, int E, int U) {
  int e = (int)(blockIdx.x * blockDim.x + threadIdx.x);
  if (e >= E) return;
  atomicAdd(&deg[eu[e]], 1.0f);
  atomicAdd(&deg[U + ei[e]], 1.0f);
}

__global__ void dinv_kernel(const float* __restrict__ deg, float* __restrict__ dinv, int N) {
  int i = (int)(blockIdx.x * blockDim.x + threadIdx.x);
  if (i >= N) return;
  float d = deg[i];
  dinv[i] = (d > 0.0f) ? rsqrtf(d) : 0.0f;
}

__global__ void zero_kernel(float* __restrict__ p, int n) {
  int i = (int)(blockIdx.x * blockDim.x + threadIdx.x);
  if (i < n) p[i] = 0.0f;
}

// ---------------------------------------------------------------------------
// LGConv scatter: one wave per edge, lane = hidden channel. Handles both
// directions (user->item, item->user) with symmetric weight w = dinv_u*dinv_i.
// 128B coalesced gathers; non-returning f32 atomics for the scatter.
// ---------------------------------------------------------------------------
__global__ void lgconv_scatter(const float* __restrict__ x, float* __restrict__ y,
                               const int* __restrict__ eu, const int* __restrict__ ei,
                               const float* __restrict__ dinv, int E, int U) {
  int wave = (int)((blockIdx.x * blockDim.x + threadIdx.x) >> 5);
  int lane = (int)(threadIdx.x & 31);
  if (wave >= E) return;
  int u = eu[wave];
  int v = U + ei[wave];
  float w = dinv[u] * dinv[v];
  float xu = x[(size_t)u * HID + lane];
  float xv = x[(size_t)v * HID + lane];
  atomicAdd(&y[(size_t)v * HID + lane], xu * w);
  atomicAdd(&y[(size_t)u * HID + lane], xv * w);
}

__global__ void finalize_kernel(const float* __restrict__ x0, const float* __restrict__ y1,
                                const float* __restrict__ y2, float* __restrict__ out, int n) {
  int i = (int)(blockIdx.x * blockDim.x + threadIdx.x);
  if (i < n) out[i] = (x0[i] + y1[i] + y2[i]) * (1.0f / 3.0f);
}

extern "C" void kernel_launch(void* const* d_in, const int* in_sizes, int n_in,
                              void* d_out, int out_size, void* d_ws, size_t ws_size,
                              hipStream_t stream) {
  const float* x_user = (const float*)d_in[0];
  const float* x_item = (const float*)d_in[1];
  const int*   eu     = (const int*)d_in[2];
  const int*   ei     = (const int*)d_in[3];
  const float* W_user = (const float*)d_in[4];
  const float* b_user = (const float*)d_in[5];
  const float* W_item = (const float*)d_in[6];
  const float* b_item = (const float*)d_in[7];
  float* out = (float*)d_out;

  const int U = in_sizes[0] / IND;
  const int I = in_sizes[1] / IND;
  const int E = in_sizes[2];
  const int N = U + I;
  const size_t NH = (size_t)N * HID;

  // Workspace layout (floats): x0 | y1 | y2 | deg | dinv
  float* ws  = (float*)d_ws;
  float* x0  = ws;
  float* y1  = x0 + NH;
  float* y2  = y1 + NH;
  float* deg = y2 + NH;
  float* dnv = deg + N;

  const int TB = 256;

  // 1) Projections (WMMA). One wave per 16x16 tile.
  {
    int tilesU = (U / 16) * (HID / 16);
    int thrU = tilesU * 32;
    proj_wmma_f32<<<(thrU + TB - 1) / TB, TB, 0, stream>>>(x_user, W_user, b_user, x0, U);
    int tilesI = (I / 16) * (HID / 16);
    int thrI = tilesI * 32;
    proj_wmma_f32<<<(thrI + TB - 1) / TB, TB, 0, stream>>>(x_item, W_item, b_item, x0 + (size_t)U * HID, I);
  }

  // 2) Degrees and symmetric normalization.
  zero_kernel<<<(N + TB - 1) / TB, TB, 0, stream>>>(deg, N);
  degree_kernel<<<(E + TB - 1) / TB, TB, 0, stream>>>(eu, ei, deg, E, U);
  dinv_kernel<<<(N + TB - 1) / TB, TB, 0, stream>>>(deg, dnv, N);

  // 3) LGConv layer 1: y1 = A_norm * x0
  zero_kernel<<<((int)NH + TB - 1) / TB, TB, 0, stream>>>(y1, (int)NH);
  {
    long long thr = (long long)E * 32;
    lgconv_scatter<<<(unsigned)((thr + TB - 1) / TB), TB, 0, stream>>>(x0, y1, eu, ei, dnv, E, U);
  }

  // 4) LGConv layer 2: y2 = A_norm * y1
  zero_kernel<<<((int)NH + TB - 1) / TB, TB, 0, stream>>>(y2, (int)NH);
  {
    long long thr = (long long)E * 32;
    lgconv_scatter<<<(unsigned)((thr + TB - 1) / TB), TB, 0, stream>>>(y1, y2, eu, ei, dnv, E, U);
  }

  // 5) Final mean over {x0, y1, y2}; node order (users then items) already
  //    matches the concatenated reference output.
  finalize_kernel<<<((int)NH + TB - 1) / TB, TB, 0, stream>>>(x0, y1, y2, out, (int)NH);
}